// GCNLayer_901943132166
// MI455X (gfx1250) — compile-verified
//
#include <hip/hip_runtime.h>

typedef float v2f __attribute__((ext_vector_type(2)));
typedef float v8f __attribute__((ext_vector_type(8)));

#define IN_FEATS 128

// ---------------- zero init: d_out (used as agg accumulator) + degree arrays ---------
__global__ void gcn_zero_kernel(float* __restrict__ out, float* __restrict__ deg,
                                int n_out4, int n_deg) {
    int i = blockIdx.x * blockDim.x + threadIdx.x;
    if (i < n_out4) ((float4*)out)[i] = make_float4(0.f, 0.f, 0.f, 0.f);
    if (i < n_deg)  deg[i] = 0.f;
}

// ---------------- degree counting: out_deg[src[e]]++, in_deg[dst[e]]++ --------------
__global__ void gcn_degree_kernel(const int* __restrict__ src, const int* __restrict__ dst,
                                  float* __restrict__ out_deg, float* __restrict__ in_deg,
                                  int E) {
    int e = blockIdx.x * blockDim.x + threadIdx.x;
    if (e >= E) return;
    atomicAdd(&out_deg[src[e]], 1.0f);
    atomicAdd(&in_deg[dst[e]], 1.0f);
}

// ---------------- edge scatter: agg[dst] += feat[src] * rsqrt(max(out_deg[src],1)) --
// One wave32 per edge: lanes cover the 128-float feature row as float4 (512B coalesced
// load), then 4x global_atomic_add_f32 per lane into the destination row.
__global__ void gcn_scatter_kernel(const float* __restrict__ feat,
                                   const int* __restrict__ src,
                                   const int* __restrict__ dst,
                                   const float* __restrict__ out_deg,
                                   float* __restrict__ agg, int E) {
    int gid  = blockIdx.x * blockDim.x + threadIdx.x;
    int e    = gid >> 5;
    int lane = gid & 31;
    if (e >= E) return;
    int s = src[e];
    int d = dst[e];
    float ns = rsqrtf(fmaxf(out_deg[s], 1.0f));
    float4 v = *(const float4*)(feat + (size_t)s * IN_FEATS + lane * 4);
    float* ap = agg + (size_t)d * IN_FEATS + lane * 4;
    atomicAdd(ap + 0, v.x * ns);
    atomicAdd(ap + 1, v.y * ns);
    atomicAdd(ap + 2, v.z * ns);
    atomicAdd(ap + 3, v.w * ns);
}

// ---------------- in-place WMMA matmul + post-scale + bias --------------------------
// out[r,:] = (out[r,:] @ W) * rsqrt(max(in_deg[r],1)) + bias
// Block = 16 rows x 128 cols; 8 waves, wave w owns the 16x16 tile at cols [16w,16w+16).
// A panel (16x128 f32) staged in LDS (stride 132 -> conflict-free across 64 banks),
// then 32 chained V_WMMA_F32_16X16X4_F32 per tile.
__global__ void __launch_bounds__(256)
gcn_matmul_kernel(float* __restrict__ out,
                  const float* __restrict__ W,
                  const float* __restrict__ bias,
                  const float* __restrict__ in_deg) {
    __shared__ float As[16][132];

    int tid  = threadIdx.x;
    int wave = tid >> 5;
    int lane = tid & 31;
    int r0   = blockIdx.x * 16;

    // cooperative load of the 16x128 A panel (agg rows) into LDS
    for (int i = tid; i < 16 * 32; i += 256) {
        int row = i >> 5;
        int c4  = (i & 31) * 4;
        float4 v = *(const float4*)(out + (size_t)(r0 + row) * IN_FEATS + c4);
        As[row][c4 + 0] = v.x;
        As[row][c4 + 1] = v.y;
        As[row][c4 + 2] = v.z;
        As[row][c4 + 3] = v.w;
    }
    __syncthreads();

    int c0    = wave * 16;
    int mlane = lane & 15;   // M for A frag / N for B frag / N for C frag
    int hi    = lane >> 4;   // lane-group selects K pair (A/B) or M+8 half (C/D)

    v8f acc = {};
    for (int k0 = 0; k0 < IN_FEATS; k0 += 4) {
        int kb = k0 + hi * 2;
        v2f a, b;
        // A 16x4 f32: lanes 0-15 hold K={k0,k0+1}, lanes 16-31 hold K={k0+2,k0+3}
        a.x = As[mlane][kb];
        a.y = As[mlane][kb + 1];
        // B 4x16 f32: N striped across lanes, K pair selected by lane group
        b.x = W[(size_t)kb * IN_FEATS + c0 + mlane];
        b.y = W[(size_t)(kb + 1) * IN_FEATS + c0 + mlane];
        acc = __builtin_amdgcn_wmma_f32_16x16x4_f32(
            /*neg_a=*/false, a, /*neg_b=*/false, b,
            /*c_mod=*/(short)0, acc, /*reuse_a=*/false, /*reuse_b=*/false);
    }

    // epilogue: C/D layout -> VGPR v holds row M=v (lanes 0-15) / M=v+8 (lanes 16-31)
    float bcol = bias[c0 + mlane];
    for (int v = 0; v < 8; ++v) {
        int row  = r0 + v + hi * 8;
        float nd = rsqrtf(fmaxf(in_deg[row], 1.0f));
        out[(size_t)row * IN_FEATS + c0 + mlane] = acc[v] * nd + bcol;
    }
}

extern "C" void kernel_launch(void* const* d_in, const int* in_sizes, int n_in,
                              void* d_out, int out_size, void* d_ws, size_t ws_size,
                              hipStream_t stream) {
    const float* feat = (const float*)d_in[0];
    const float* W    = (const float*)d_in[1];
    const float* bias = (const float*)d_in[2];
    const int*   src  = (const int*)d_in[3];
    const int*   dst  = (const int*)d_in[4];
    float* out = (float*)d_out;

    int N = in_sizes[0] / IN_FEATS;   // 100000
    int E = in_sizes[3];              // 1600000

    float* out_deg = (float*)d_ws;    // N floats
    float* in_deg  = out_deg + N;     // N floats  (ws usage: 2*N*4 = 800 KB)

    int n_out4 = N * (IN_FEATS / 4);
    int n_deg  = 2 * N;
    int zmax   = (n_out4 > n_deg) ? n_out4 : n_deg;

    gcn_zero_kernel<<<(zmax + 255) / 256, 256, 0, stream>>>(out, out_deg, n_out4, n_deg);
    gcn_degree_kernel<<<(E + 255) / 256, 256, 0, stream>>>(src, dst, out_deg, in_deg, E);
    gcn_scatter_kernel<<<(E + 7) / 8, 256, 0, stream>>>(feat, src, dst, out_deg, out, E);
    gcn_matmul_kernel<<<N / 16, 256, 0, stream>>>(out, W, bias, in_deg);
}